// LinearAppearance_83476984365255
// MI455X (gfx1250) — compile-verified
//
#include <hip/hip_runtime.h>

// Problem dims (fixed by the reference): B=8, N=128, T=64, F=1799, D=256
constexpr int Bb = 8, Nn = 128, Tt = 64, Ff = 1799, Dd = 256;
constexpr int BN = Bb * Nn;           // 1024 (b,n) rows
constexpr float ALPHA = 0.9f;
constexpr float ONE_MINUS_ALPHA = 0.1f;

typedef __attribute__((ext_vector_type(2))) float v2f;
typedef __attribute__((ext_vector_type(8))) float v8f;

// ---------------------------------------------------------------------------
// Kernel 1: fold the EMA over T into the feature vectors.
//   g[bn,f] = sum_t w[bn,t] * feats[bn,t,f]
//   sw[bn]  = sum_t w[bn,t]   (bias coefficient = 1 - (1-a)^K)
// One block per (b,n); 64x1799 fp32 streamed once with NT hint (read-once,
// 472MB total > 192MB L2, so non-temporal avoids L2 thrash). Zero-weight
// timesteps (mask==0) are skipped uniformly -> ~half the HBM traffic.
// ---------------------------------------------------------------------------
__global__ void __launch_bounds__(256)
ema_fold_kernel(const float* __restrict__ feats,
                const unsigned char* __restrict__ masks,
                float* __restrict__ g,
                float* __restrict__ sw) {
  __shared__ float wLds[Tt];
  __shared__ float swLds;

  const int bn  = blockIdx.x;
  const int tid = threadIdx.x;

  if (tid == 0) {
    // backward scan: w_t = mask_t ? a * (1-a)^(#masked after t) : 0
    float prod = 1.0f;
    const unsigned char* m = masks + (size_t)bn * Tt;
    for (int t = Tt - 1; t >= 0; --t) {
      if (m[t]) { wLds[t] = ALPHA * prod; prod *= ONE_MINUS_ALPHA; }
      else      { wLds[t] = 0.0f; }
    }
    swLds = 1.0f - prod;  // sum of weights
  }
  __syncthreads();

  float acc[8];
#pragma unroll
  for (int j = 0; j < 8; ++j) acc[j] = 0.0f;

  const float* base = feats + (size_t)bn * Tt * Ff;
  for (int t = 0; t < Tt; ++t) {
    const float wt = wLds[t];
    if (wt == 0.0f) continue;  // uniform across block: no divergence
    const float* row = base + (size_t)t * Ff;
#pragma unroll
    for (int j = 0; j < 8; ++j) {
      const int f = tid + j * 256;
      if (f < Ff) acc[j] += wt * __builtin_nontemporal_load(row + f);
    }
  }

  float* grow = g + (size_t)bn * Ff;
#pragma unroll
  for (int j = 0; j < 8; ++j) {
    const int f = tid + j * 256;
    if (f < Ff) grow[f] = acc[j];
  }
  if (tid == 0) sw[bn] = swLds;
}

// ---------------------------------------------------------------------------
// Kernel 2: ema[r,d] = g[r,:] . W[d,:] + sw[r] * bias[d]
// One wave32 per 16x16 output tile, fp32 WMMA 16x16x4, K-loop over F=1799.
// A (16x4 fp32) lane layout: lane = m (mod 16); v[0]=K(half*2), v[1]=K(half*2+1)
// B (4x16  fp32) lane layout: lane = n (mod 16); same K striping as A.
// C/D (16x16 fp32): VGPR j, lanes 0-15 -> M=j, lanes 16-31 -> M=j+8.
// ---------------------------------------------------------------------------
__global__ void __launch_bounds__(32)
wmma_gemm_kernel(const float* __restrict__ g,
                 const float* __restrict__ W,
                 const float* __restrict__ bias,
                 const float* __restrict__ sw,
                 float* __restrict__ out) {
  const int lane    = threadIdx.x;      // 0..31, one wave32
  const int half    = lane >> 4;        // 0: K+0/1, 1: K+2/3
  const int l16     = lane & 15;
  const int rowBase = blockIdx.x * 16;  // over BN = 1024
  const int colBase = blockIdx.y * 16;  // over D = 256

  // Per-lane base pointers (include the half*2 K-offset of this lane).
  const float* arow = g + (size_t)(rowBase + l16) * Ff + half * 2;  // A[m, k]
  const float* brow = W + (size_t)(colBase + l16) * Ff + half * 2;  // B[k, n] = W[n, k]

  v8f c = {};  // fp32 accumulator, 8 VGPRs

  // Main K loop: 449 iterations cover f = 0..1795.
  int k = 0;
  for (; k + 4 <= Ff - 3; k += 4) {
    v2f a, b;
    a.x = arow[k];
    a.y = arow[k + 1];
    b.x = brow[k];
    b.y = brow[k + 1];
    c = __builtin_amdgcn_wmma_f32_16x16x4_f32(
        /*neg_a=*/false, a, /*neg_b=*/false, b,
        /*c_mod=*/(short)0, c, /*reuse_a=*/false, /*reuse_b=*/false);
  }

  // Tail: f = 1796,1797,1798 valid; K=1799 zero-filled (lane-half 1, v[1]).
  {
    v2f a, b;
    a.x = arow[k];                                // K=1796 or 1798: valid
    a.y = half ? 0.0f : arow[k + 1];              // K=1797 valid, K=1799 OOB
    b.x = brow[k];
    b.y = half ? 0.0f : brow[k + 1];
    c = __builtin_amdgcn_wmma_f32_16x16x4_f32(
        false, a, false, b, (short)0, c, false, false);
  }

  // Epilogue: ema[r,d] = c + sw[r] * bias[d]
  const int d = colBase + l16;
  const float bd = bias[d];
#pragma unroll
  for (int j = 0; j < 8; ++j) {
    const int r = rowBase + j + half * 8;
    out[(size_t)r * Dd + d] = c[j] + sw[r] * bd;
  }
}

// ---------------------------------------------------------------------------
extern "C" void kernel_launch(void* const* d_in, const int* in_sizes, int n_in,
                              void* d_out, int out_size, void* d_ws, size_t ws_size,
                              hipStream_t stream) {
  (void)in_sizes; (void)n_in; (void)out_size; (void)ws_size;

  const float*         feats = (const float*)d_in[0];          // [8,128,64,1799]
  const unsigned char* masks = (const unsigned char*)d_in[1];  // [8,128,64] bool
  const float*         W     = (const float*)d_in[2];          // [256,1799]
  const float*         bias  = (const float*)d_in[3];          // [256]
  float*               out   = (float*)d_out;                  // [1024,256]

  // Workspace layout: sw[1024] then g[1024*1799]  (~7.4 MB total)
  float* sw = (float*)d_ws;
  float* g  = sw + BN;

  ema_fold_kernel<<<dim3(BN), dim3(256), 0, stream>>>(feats, masks, g, sw);

  dim3 gridC(BN / 16, Dd / 16);  // 64 x 16 tiles
  wmma_gemm_kernel<<<gridC, dim3(32), 0, stream>>>(g, W, bias, sw, out);
}